// Hungarian_70145405878794
// MI455X (gfx1250) — compile-verified
//
#include <hip/hip_runtime.h>
#include <math.h>

typedef __attribute__((ext_vector_type(2))) float v2f;
typedef __attribute__((ext_vector_type(8))) float v8f;

#define BB 16
#define NN 512   // seq1 rows
#define MM 448   // seq2 rows
#define DD 1536

// ---------------------------------------------------------------------------
// Kernel 1: squared row norms of seq1 and seq2. One wave32 per row.
// ---------------------------------------------------------------------------
__global__ void norms_kernel(const float* __restrict__ s1,
                             const float* __restrict__ s2,
                             float* __restrict__ n1,   // [BB*NN]
                             float* __restrict__ n2)   // [BB*MM]
{
    int gwave = (blockIdx.x * blockDim.x + threadIdx.x) >> 5;
    int lane  = threadIdx.x & 31;
    const int totalRows = BB * NN + BB * MM;
    if (gwave >= totalRows) return;

    const float* src;
    float* dst;
    if (gwave < BB * NN) {
        src = s1 + (size_t)gwave * DD;
        dst = n1 + gwave;
    } else {
        int g = gwave - BB * NN;
        src = s2 + (size_t)g * DD;
        dst = n2 + g;
    }
    const float4* src4 = (const float4*)src;
    float s = 0.0f;
    for (int d4 = lane; d4 < DD / 4; d4 += 32) {
        float4 x = src4[d4];
        s += x.x * x.x + x.y * x.y + x.z * x.z + x.w * x.w;
    }
    #pragma unroll
    for (int o = 16; o > 0; o >>= 1) s += __shfl_xor(s, o, 32);
    if (lane == 0) *dst = s;
}

// ---------------------------------------------------------------------------
// Kernel 2: cost^T via WMMA fp32 GEMM + fused norm/sqrt epilogue.
// costT[b][i][j] = || seq2[b,i,:] - seq1[b,j,:] ||_2, i in [0,448), j in [0,512)
// Block tile 64(i) x 128(j); 8 waves as 2x4, each wave owns a 32x32 sub-tile
// (2 A frags x 2 B frags -> 4 V_WMMA_F32_16X16X4_F32 per K-step of 4,
//  32 B of DS reads per lane per 8192 FLOP).
// ---------------------------------------------------------------------------
#define KC 36  // padded LDS row stride (floats): 144 B, 16B-aligned

__global__ __launch_bounds__(256)
void cost_wmma_kernel(const float* __restrict__ seq1,
                      const float* __restrict__ seq2,
                      const float* __restrict__ n1,    // [BB*NN]
                      const float* __restrict__ n2,    // [BB*MM]
                      float* __restrict__ costT)       // [BB][MM][NN]
{
    __shared__ float As[64][KC];    // seq2 tile (i-rows)
    __shared__ float Bs[128][KC];   // seq1 tile (j-rows)

    const int b     = blockIdx.z;
    const int iBase = blockIdx.x * 64;    // 448/64  = 7 blocks
    const int jBase = blockIdx.y * 128;   // 512/128 = 4 blocks

    const float* s2b = seq2 + (size_t)b * MM * DD;
    const float* s1b = seq1 + (size_t)b * NN * DD;

    const int tid  = threadIdx.x;
    const int lane = tid & 31;
    const int wave = tid >> 5;
    const int wr   = wave & 1;   // 2 row groups of 32
    const int wc   = wave >> 1;  // 4 col groups of 32
    const int half = lane >> 4;
    const int lr   = lane & 15;

    v8f acc[2][2] = {};

    const int col4 = (tid & 7) * 4;  // 0..28
    const int rowp = tid >> 3;       // 0..31

    for (int k0 = 0; k0 < DD; k0 += 32) {
        #pragma unroll
        for (int h = 0; h < 2; ++h) {
            int row = rowp + h * 32;
            *(float4*)&As[row][col4] =
                *(const float4*)(s2b + (size_t)(iBase + row) * DD + k0 + col4);
        }
        #pragma unroll
        for (int h = 0; h < 4; ++h) {
            int row = rowp + h * 32;
            *(float4*)&Bs[row][col4] =
                *(const float4*)(s1b + (size_t)(jBase + row) * DD + k0 + col4);
        }
        __syncthreads();

        #pragma unroll
        for (int kk = 0; kk < 32; kk += 4) {
            int ka = kk + 2 * half;
            v2f a0 = *(const v2f*)&As[wr * 32 + lr][ka];
            v2f a1 = *(const v2f*)&As[wr * 32 + 16 + lr][ka];
            v2f b0 = *(const v2f*)&Bs[wc * 32 + lr][ka];
            v2f b1 = *(const v2f*)&Bs[wc * 32 + 16 + lr][ka];
            acc[0][0] = __builtin_amdgcn_wmma_f32_16x16x4_f32(
                false, a0, false, b0, (short)0, acc[0][0], false, false);
            acc[0][1] = __builtin_amdgcn_wmma_f32_16x16x4_f32(
                false, a0, false, b1, (short)0, acc[0][1], false, false);
            acc[1][0] = __builtin_amdgcn_wmma_f32_16x16x4_f32(
                false, a1, false, b0, (short)0, acc[1][0], false, false);
            acc[1][1] = __builtin_amdgcn_wmma_f32_16x16x4_f32(
                false, a1, false, b1, (short)0, acc[1][1], false, false);
        }
        __syncthreads();
    }

    // epilogue: cost = sqrt(max(|s2_i|^2 + |s1_j|^2 - 2*dot, 0)), stored at
    // costT[b][i][j] (transposed vs the reference's [N][M] cost so the
    // Hungarian row scans are coalesced).
    const float* n1b = n1 + b * NN;
    const float* n2b = n2 + b * MM;
    const int jCol = jBase + wc * 32 + lr;
    const float bn0 = n1b[jCol];
    const float bn1 = n1b[jCol + 16];
    float* cT = costT + (size_t)b * MM * NN;

    #pragma unroll
    for (int ti = 0; ti < 2; ++ti) {
        #pragma unroll
        for (int r = 0; r < 8; ++r) {
            int iRow = iBase + wr * 32 + ti * 16 + 8 * half + r;
            float an = n2b[iRow];
            float d0 = sqrtf(fmaxf(an + bn0 - 2.0f * acc[ti][0][r], 0.0f));
            float d1 = sqrtf(fmaxf(an + bn1 - 2.0f * acc[ti][1][r], 0.0f));
            cT[(size_t)iRow * NN + jCol]      = d0;
            cT[(size_t)iRow * NN + jCol + 16] = d1;
        }
    }
}

// ---------------------------------------------------------------------------
// Kernel 3: Jonker-Volgenant LSA, one block per batch element.
// Transposed problem (matches reference, which transposes since N > M):
// n=448 rows (seq2), m=512 cols (seq1). Thread tid owns column j = tid+1;
// minv[j] and v[j] live in registers (owner-only access); argmin is a
// wave32 shfl butterfly + 16-entry cross-wave reduce -> 4 barriers/step.
// Emits perm[b][0..511]: matched seq1 rows ascending, then unmatched.
// ---------------------------------------------------------------------------
__global__ __launch_bounds__(512)
void hungarian_kernel(const float* __restrict__ costT,  // [BB][MM][NN]
                      int* __restrict__ perm)           // [BB][NN]
{
    const int n = MM;  // 448
    const int m = NN;  // 512
    const float INF = 1e30f;

    __shared__ float u[MM + 1];        // 449
    __shared__ int   p[NN + 1];        // 513
    __shared__ int   way[NN + 1];
    __shared__ int   used[NN + 1];
    __shared__ float wv[16];
    __shared__ int   wi[16];
    __shared__ int   s_j0;
    __shared__ float s_delta;
    __shared__ int   s_j1;

    const int b    = blockIdx.x;
    const int tid  = threadIdx.x;
    const int bs   = blockDim.x;     // 512
    const int lane = tid & 31;
    const int wid  = tid >> 5;       // 0..15
    const int j    = tid + 1;        // owned column, 1..512
    const float* Cb = costT + (size_t)b * MM * NN;

    float v_r = 0.0f;     // v[j], owner-only
    float minv_r;         // minv[j], owner-only

    for (int idx = tid; idx <= n; idx += bs) u[idx] = 0.0f;
    for (int idx = tid; idx <= m; idx += bs) { p[idx] = 0; way[idx] = 0; }
    __syncthreads();

    for (int i = 1; i <= n; ++i) {
        minv_r = INF;
        for (int idx = tid; idx <= m; idx += bs) used[idx] = 0;
        if (tid == 0) { p[0] = i; s_j0 = 0; }
        __syncthreads();

        while (true) {
            int j0 = s_j0;
            if (tid == 0) used[j0] = 1;
            __syncthreads();                       // barrier 1

            int   i0  = p[j0];
            float ui0 = u[i0];
            int   usd = used[j];
            if (!usd) {
                float cur = Cb[(size_t)(i0 - 1) * NN + (j - 1)] - ui0 - v_r;
                if (cur < minv_r) { minv_r = cur; way[j] = j0; }
            }

            // argmin over non-used columns (first-occurrence tie-break)
            float val  = usd ? INF : minv_r;
            int   vidx = j;
            #pragma unroll
            for (int o = 16; o > 0; o >>= 1) {
                float ov = __shfl_xor(val, o, 32);
                int   oi = __shfl_xor(vidx, o, 32);
                if (ov < val || (ov == val && oi < vidx)) { val = ov; vidx = oi; }
            }
            if (lane == 0) { wv[wid] = val; wi[wid] = vidx; }
            __syncthreads();                       // barrier 2
            if (wid == 0) {
                float v2 = (lane < 16) ? wv[lane] : INF;
                int   i2 = (lane < 16) ? wi[lane] : 0x7fffffff;
                #pragma unroll
                for (int o = 8; o > 0; o >>= 1) {
                    float ov = __shfl_xor(v2, o, 32);
                    int   oi = __shfl_xor(i2, o, 32);
                    if (ov < v2 || (ov == v2 && oi < i2)) { v2 = ov; i2 = oi; }
                }
                if (lane == 0) { s_delta = v2; s_j1 = i2; }
            }
            __syncthreads();                       // barrier 3

            float delta = s_delta;
            int   j1    = s_j1;
            if (usd) {
                u[p[j]] += delta;   // p distinct over used columns: race-free
                v_r     -= delta;
            } else {
                minv_r  -= delta;
            }
            if (tid == 0) {
                u[p[0]] += delta;   // column 0 is always in the used set here
                s_j0 = j1;
            }
            __syncthreads();                       // barrier 4
            if (p[j1] == 0) break;
        }

        if (tid == 0) {  // augmenting-path rewind (serial)
            int j0 = s_j0;
            while (j0) { int jw = way[j0]; p[j0] = p[jw]; j0 = jw; }
        }
        __syncthreads();
    }

    if (tid == 0) {
        int* pb = perm + b * NN;
        int pos = 0;
        for (int jj = 1; jj <= m; ++jj) if (p[jj] != 0) pb[pos++] = jj - 1;
        for (int jj = 1; jj <= m; ++jj) if (p[jj] == 0) pb[pos++] = jj - 1;
    }
}

// ---------------------------------------------------------------------------
// Kernel 4: out[b,k,:] = seq1[b, perm[b,k], :]
// ---------------------------------------------------------------------------
__global__ void gather_kernel(const float* __restrict__ seq1,
                              const int* __restrict__ perm,
                              float* __restrict__ out)
{
    const int k = blockIdx.x;
    const int b = blockIdx.y;
    const int src = perm[b * NN + k];
    const float4* in4  = (const float4*)(seq1 + ((size_t)b * NN + src) * DD);
    float4*       out4 = (float4*)(out + ((size_t)b * NN + k) * DD);
    for (int d = threadIdx.x; d < DD / 4; d += blockDim.x) out4[d] = in4[d];
}

// ---------------------------------------------------------------------------
extern "C" void kernel_launch(void* const* d_in, const int* in_sizes, int n_in,
                              void* d_out, int out_size, void* d_ws, size_t ws_size,
                              hipStream_t stream) {
    (void)in_sizes; (void)n_in; (void)out_size; (void)ws_size;
    const float* seq1 = (const float*)d_in[0];
    const float* seq2 = (const float*)d_in[1];
    float* out = (float*)d_out;

    float* costT = (float*)d_ws;                       // 16*448*512 f32
    float* n1    = costT + (size_t)BB * MM * NN;       // 16*512 f32
    float* n2    = n1 + BB * NN;                       // 16*448 f32
    int*   perm  = (int*)(n2 + BB * MM);               // 16*512 i32

    {
        int totalRows = BB * NN + BB * MM;
        int blocks = (totalRows * 32 + 255) / 256;
        norms_kernel<<<blocks, 256, 0, stream>>>(seq1, seq2, n1, n2);
    }
    {
        dim3 grid(MM / 64, NN / 128, BB);  // (7, 4, 16)
        cost_wmma_kernel<<<grid, 256, 0, stream>>>(seq1, seq2, n1, n2, costT);
    }
    hungarian_kernel<<<BB, 512, 0, stream>>>(costT, perm);
    {
        dim3 grid(NN, BB);
        gather_kernel<<<grid, 256, 0, stream>>>(seq1, perm, out);
    }
}